// PathModule_42838003810951
// MI455X (gfx1250) — compile-verified
//
#include <hip/hip_runtime.h>
#include <math.h>

typedef float v2f __attribute__((ext_vector_type(2)));
typedef float v8f __attribute__((ext_vector_type(8)));

#define B_SZ 256
#define N_SZ 384
#define D_SZ 128
#define CLIPP 10.0f
#define NEGV -1e9f
#define SCALE 0.25f  // 1/sqrt(16)

// ---------------------------------------------------------------------------
// Kernel 1: kv precompute GEMM via V_WMMA_F32_16X16X4_F32.
// kvT[b][j][n] = sum_k hvec[b][n][k] * kv_p[k][j]
// One wave computes a full 16(M)x128(J) strip: 8 accumulators, A fragment
// loaded once per k-step and reused across 8 back-to-back WMMAs.
// Rows are flat r = b*384+n; 384 % 16 == 0 so strips never cross a batch,
// making bidx/node0 per-wave constants (no per-element division).
// ---------------------------------------------------------------------------
__global__ void kv_gemm_wmma(const float* __restrict__ hvec,
                             const float* __restrict__ kv_p,
                             float* __restrict__ kvT) {
    const int lane  = threadIdx.x & 31;
    const int wave  = threadIdx.x >> 5;
    const int strip = blockIdx.x * 8 + wave;     // 6144 strips total
    const int row0  = strip * 16;
    const int bidx  = row0 / N_SZ;               // uniform per wave
    const int node0 = row0 - bidx * N_SZ;

    // A 16x4 f32 layout: lane L holds M=L%16; VGPR0=K=koff, VGPR1=K=koff+1,
    // koff=(L/16)*2.  B 4x16 mirrors: N=L%16, same K split.
    const int m16  = lane & 15;
    const int koff = (lane >> 4) * 2;

    v8f acc[8] = {};
    const float* aptr = hvec + (size_t)(row0 + m16) * D_SZ + koff;

    for (int k0 = 0; k0 < D_SZ; k0 += 4) {
        v2f a = *(const v2f*)(aptr + k0);
        const float* bptr = kv_p + (size_t)(k0 + koff) * D_SZ + m16;
#pragma unroll
        for (int j = 0; j < 8; ++j) {
            v2f bm;
            bm.x = bptr[j * 16];
            bm.y = bptr[D_SZ + j * 16];
            acc[j] = __builtin_amdgcn_wmma_f32_16x16x4_f32(
                /*neg_a=*/false, a, /*neg_b=*/false, bm,
                /*c_mod=*/(short)0, acc[j], /*reuse_a=*/false, /*reuse_b=*/false);
        }
    }

    // C/D 16x16 f32 layout: lane L col = L%16, VGPR r -> M = r + 8*(L/16).
    const int ncol  = lane & 15;
    const int mbase = (lane >> 4) * 8;
    float* obase = kvT + (size_t)bidx * D_SZ * N_SZ + (node0 + mbase);
#pragma unroll
    for (int j = 0; j < 8; ++j) {
        float* oj = obase + (size_t)(j * 16 + ncol) * N_SZ;
#pragma unroll
        for (int r = 0; r < 8; ++r)
            oj[r] = acc[j][r];
    }
}

// ---------------------------------------------------------------------------
// Kernel 2: full 384-step greedy decode. One block per batch element
// (carry is per-batch independent), 384 threads = 12 waves, thread n owns
// node n.  kv[b] (196 KB) streams from L2 each step, coalesced via kvT.
// ---------------------------------------------------------------------------
__global__ void __launch_bounds__(N_SZ)
decode_kernel(const float* __restrict__ hvec,
              const float* __restrict__ hbar,
              const float* __restrict__ qv_p,
              const float* __restrict__ vec1,
              const float* __restrict__ vecf,
              const float* __restrict__ kvT,
              float* __restrict__ out) {
    const int b   = blockIdx.x;
    const int tid = threadIdx.x;

    __shared__ float s_last[D_SZ];
    __shared__ float s_first[D_SZ];
    __shared__ float s_ctx[D_SZ];
    __shared__ float s_q[D_SZ];
    __shared__ float s_red[512];
    __shared__ int   s_idx[512];
    __shared__ unsigned char s_vis[N_SZ];
    __shared__ int   s_a;
    __shared__ float s_max;

    if (tid < D_SZ) { s_last[tid] = vec1[tid]; s_first[tid] = vecf[tid]; }
    s_vis[tid] = 0;
    float logp = 0.0f;
    __syncthreads();

    const float* kvb = kvT + (size_t)b * D_SZ * N_SZ;
    const float* hb  = hbar + (size_t)b * D_SZ;
    const float* hv  = hvec + (size_t)b * N_SZ * D_SZ;

    for (int t = 0; t < N_SZ; ++t) {
        // ctx = hbar + last + first
        if (tid < D_SZ) s_ctx[tid] = hb[tid] + s_last[tid] + s_first[tid];
        __syncthreads();

        // q[j] = sum_k qv_p[k][j] * ctx[k]   (coalesced over j)
        if (tid < D_SZ) {
            float acc = 0.0f;
#pragma unroll 4
            for (int k = 0; k < D_SZ; ++k)
                acc += qv_p[(size_t)k * D_SZ + tid] * s_ctx[k];
            s_q[tid] = acc;
        }
        __syncthreads();

        // scores[n] = sum_k kvT[b][k][n] * q[k]   (coalesced over n)
        float acc = 0.0f;
#pragma unroll 8
        for (int k = 0; k < D_SZ; ++k)
            acc += kvb[(size_t)k * N_SZ + tid] * s_q[k];
        float score  = CLIPP * tanhf(acc * SCALE);
        float masked = s_vis[tid] ? NEGV : score;

        // --- argmax (first-index tie-break, matches jnp.argmax) ---
        s_red[tid] = masked;
        s_idx[tid] = tid;
        if (tid < 128) { s_red[N_SZ + tid] = -3.0e38f; s_idx[N_SZ + tid] = 0x7fffffff; }
        __syncthreads();
        for (int s = 256; s > 0; s >>= 1) {
            if (tid < s) {
                float v2 = s_red[tid + s]; int i2 = s_idx[tid + s];
                float v1 = s_red[tid];     int i1 = s_idx[tid];
                if (v2 > v1 || (v2 == v1 && i2 < i1)) { s_red[tid] = v2; s_idx[tid] = i2; }
            }
            __syncthreads();
        }
        if (tid == 0) { s_a = s_idx[0]; s_max = s_red[0]; }
        __syncthreads();

        // --- log-sum-exp ---
        s_red[tid] = expf(masked - s_max);
        if (tid < 128) s_red[N_SZ + tid] = 0.0f;
        __syncthreads();
        for (int s = 256; s > 0; s >>= 1) {
            if (tid < s) s_red[tid] += s_red[tid + s];
            __syncthreads();
        }

        const int a = s_a;
        if (tid == 0) {
            // logpol[a] = masked[a] - (max + log(sumexp)) = -log(sumexp)
            logp -= logf(s_red[0]);
            out[(size_t)t * B_SZ + b] = (float)a;   // pi[t][b]
            s_vis[a] = 1;
        }
        __syncthreads();

        if (tid < D_SZ) {
            float ch = hv[(size_t)a * D_SZ + tid];
            s_last[tid] = ch;
            if (t == 0) s_first[tid] = ch;
        }
        __syncthreads();
    }

    if (tid == 0) out[(size_t)N_SZ * B_SZ + b] = logp;
}

extern "C" void kernel_launch(void* const* d_in, const int* in_sizes, int n_in,
                              void* d_out, int out_size, void* d_ws, size_t ws_size,
                              hipStream_t stream) {
    const float* hvec = (const float*)d_in[0];  // [256,384,128]
    const float* hbar = (const float*)d_in[1];  // [256,128]
    const float* qv_p = (const float*)d_in[2];  // [128,8,16] -> [128,128]
    const float* kv_p = (const float*)d_in[3];  // [128,8,16] -> [128,128]
    const float* vec1 = (const float*)d_in[4];  // [128]
    const float* vecf = (const float*)d_in[5];  // [128]
    float* out = (float*)d_out;                 // pi [384,256] ++ logp [256]

    float* kvT = (float*)d_ws;                  // [256][128][384] = 50.3 MB

    // Phase 1: WMMA GEMM. 6144 strips (16 rows x 128 cols), 8 waves/block.
    kv_gemm_wmma<<<768, 256, 0, stream>>>(hvec, kv_p, kvT);

    // Phase 2: batch-parallel sequential decode, one block per batch element.
    decode_kernel<<<B_SZ, N_SZ, 0, stream>>>(hvec, hbar, qv_p, vec1, vecf, kvT, out);
}